// GraphSage_42494406427538
// MI455X (gfx1250) — compile-verified
//
#include <hip/hip_runtime.h>

// ---------------------------------------------------------------------------
// GraphSAGE (pool aggregator) on MI455X / gfx1250, fp32 end-to-end.
// GEMMs use V_WMMA_F32_16X16X4_F32 (wave32, 16x16 f32 tile per wave).
// segment_max implemented as u32 atomic max (valid: operands are >= 0).
// ---------------------------------------------------------------------------

typedef float v2f __attribute__((ext_vector_type(2)));
typedef float v8f __attribute__((ext_vector_type(8)));

#define N_NODES 100000
#define N_EDGES 1600000
#define KDIM    128      // IN_F == HID_F == 128

__device__ __forceinline__ v8f wmma_f32(v2f a, v2f b, v8f c) {
    // 8 args: (neg_a, A, neg_b, B, c_mod, C, reuse_a, reuse_b)
    return __builtin_amdgcn_wmma_f32_16x16x4_f32(
        false, a, false, b, (short)0, c, false, false);
}

// ---------------------------------------------------------------------------
// Y[N x 128] = relu(X[N x 128] @ W[128 x 128]^T + bias)
// 1 block = 16 rows; 8 waves, wave w computes the 16x16 tile at cols w*16.
// A tile staged in LDS once per block.
// ---------------------------------------------------------------------------
__global__ void pool_gemm_relu(const float* __restrict__ X,
                               const float* __restrict__ W,
                               const float* __restrict__ bias,
                               float* __restrict__ Y) {
    __shared__ float As[16 * KDIM];
    const int tid      = threadIdx.x;
    const int row_base = blockIdx.x * 16;

    {   // stage 16x128 A tile (2048 floats = 512 float4)
        const float4* s4 = (const float4*)(X + (size_t)row_base * KDIM);
        float4* d4 = (float4*)As;
        for (int i = tid; i < 512; i += blockDim.x) d4[i] = s4[i];
    }
    __syncthreads();

    const int wave = tid >> 5;
    const int lane = tid & 31;
    const int m    = lane & 15;          // row within tile (A/C) & col (B/C)
    const int kh   = (lane >> 4) << 1;   // K sub-offset: lanes 0-15 -> {0,1}, 16-31 -> {2,3}
    const int col  = wave * 16 + m;

    const float* arow = As + m * KDIM + kh;
    const float* brow = W + (size_t)col * KDIM + kh;   // B[k][n] = W[n][k]

    v8f c = {};
#pragma unroll
    for (int k0 = 0; k0 < KDIM; k0 += 4) {
        v2f a = { arow[k0], arow[k0 + 1] };
        v2f b = { brow[k0], brow[k0 + 1] };
        c = wmma_f32(a, b, c);
    }

    const float bv = bias[col];
#pragma unroll
    for (int r = 0; r < 8; ++r) {
        const int orow = row_base + r + ((lane >> 4) << 3);  // M = r or r+8
        float v = c[r] + bv;
        v = fmaxf(v, 0.0f);
        Y[(size_t)orow * KDIM + col] = v;
    }
}

// ---------------------------------------------------------------------------
// Y[N x O] = act(X1 @ Wself^T + bself + X2 @ Wneigh^T + bneigh)
// nWaves = ceil(O/16) waves per block; column-masked for O=40.
// ---------------------------------------------------------------------------
template <bool RELU>
__global__ void combine_gemm(const float* __restrict__ X1,
                             const float* __restrict__ Wself,
                             const float* __restrict__ bself,
                             const float* __restrict__ X2,
                             const float* __restrict__ Wneigh,
                             const float* __restrict__ bneigh,
                             float* __restrict__ Y, int O) {
    __shared__ float As1[16 * KDIM];
    __shared__ float As2[16 * KDIM];
    const int tid      = threadIdx.x;
    const int row_base = blockIdx.x * 16;

    {
        const float4* s1 = (const float4*)(X1 + (size_t)row_base * KDIM);
        const float4* s2 = (const float4*)(X2 + (size_t)row_base * KDIM);
        float4* d1 = (float4*)As1;
        float4* d2 = (float4*)As2;
        for (int i = tid; i < 512; i += blockDim.x) { d1[i] = s1[i]; d2[i] = s2[i]; }
    }
    __syncthreads();

    const int wave  = tid >> 5;
    const int lane  = tid & 31;
    const int m     = lane & 15;
    const int kh    = (lane >> 4) << 1;
    const int col   = wave * 16 + m;
    const bool valid = (col < O);
    const int colW  = valid ? col : (O - 1);   // clamp: keep loads in-bounds

    const float* a1 = As1 + m * KDIM + kh;
    const float* a2 = As2 + m * KDIM + kh;
    const float* b1 = Wself  + (size_t)colW * KDIM + kh;
    const float* b2 = Wneigh + (size_t)colW * KDIM + kh;

    v8f c = {};
#pragma unroll
    for (int k0 = 0; k0 < KDIM; k0 += 4) {
        v2f av = { a1[k0], a1[k0 + 1] };
        v2f bv = { b1[k0], b1[k0 + 1] };
        c = wmma_f32(av, bv, c);
    }
#pragma unroll
    for (int k0 = 0; k0 < KDIM; k0 += 4) {
        v2f av = { a2[k0], a2[k0 + 1] };
        v2f bv = { b2[k0], b2[k0 + 1] };
        c = wmma_f32(av, bv, c);
    }

    const float bb = bself[colW] + bneigh[colW];
#pragma unroll
    for (int r = 0; r < 8; ++r) {
        const int orow = row_base + r + ((lane >> 4) << 3);
        float v = c[r] + bb;
        if (RELU) v = fmaxf(v, 0.0f);
        if (valid) Y[(size_t)orow * O + col] = v;
    }
}

// ---------------------------------------------------------------------------
// Per-edge scatter-max: one wave per edge, lane handles 4 features.
// m values are post-ReLU (>= 0) so IEEE order == u32 order -> native
// non-returning GLOBAL_ATOMIC_MAX_U32.
// ---------------------------------------------------------------------------
__global__ void edge_max(const float* __restrict__ m,
                         const int* __restrict__ src,
                         const int* __restrict__ dst,
                         unsigned int* __restrict__ agg) {
    const int idx = blockIdx.x * blockDim.x + threadIdx.x;   // < E*32 = 51.2M
    const int e   = idx >> 5;
    if (e >= N_EDGES) return;
    const int c4  = (idx & 31) << 2;

    const int s = src[e];
    const int d = dst[e];
    const float4 v = *(const float4*)(m + (size_t)s * KDIM + c4);
    unsigned int* p = agg + (size_t)d * KDIM + c4;
    atomicMax(p + 0, __float_as_uint(v.x));
    atomicMax(p + 1, __float_as_uint(v.y));
    atomicMax(p + 2, __float_as_uint(v.z));
    atomicMax(p + 3, __float_as_uint(v.w));
}

// ---------------------------------------------------------------------------
extern "C" void kernel_launch(void* const* d_in, const int* in_sizes, int n_in,
                              void* d_out, int out_size, void* d_ws, size_t ws_size,
                              hipStream_t stream) {
    const float* in_feat  = (const float*)d_in[0];
    const int*   src      = (const int*)d_in[1];
    const int*   dst      = (const int*)d_in[2];
    const float* pool_w1  = (const float*)d_in[3];
    const float* pool_b1  = (const float*)d_in[4];
    const float* self_w1  = (const float*)d_in[5];
    const float* self_b1  = (const float*)d_in[6];
    const float* neigh_w1 = (const float*)d_in[7];
    const float* neigh_b1 = (const float*)d_in[8];
    const float* pool_w2  = (const float*)d_in[9];
    const float* pool_b2  = (const float*)d_in[10];
    const float* self_w2  = (const float*)d_in[11];
    const float* self_b2  = (const float*)d_in[12];
    const float* neigh_w2 = (const float*)d_in[13];
    const float* neigh_b2 = (const float*)d_in[14];
    float* out = (float*)d_out;

    const size_t NODE_BYTES = (size_t)N_NODES * KDIM * sizeof(float); // 51.2 MB
    char* ws = (char*)d_ws;
    float* m_buf = (float*)(ws);                    // pooled messages
    float* agg   = (float*)(ws + NODE_BYTES);       // segment-max result
    float* h1    = (float*)(ws + 2 * NODE_BYTES);   // layer-1 hidden

    const int nTiles     = N_NODES / 16;            // 6250 (exact)
    const int edgeBlocks = (N_EDGES * 32 + 255) / 256;

    // ---- Layer 1 ----
    hipMemsetAsync(agg, 0, NODE_BYTES, stream);
    pool_gemm_relu<<<nTiles, 256, 0, stream>>>(in_feat, pool_w1, pool_b1, m_buf);
    edge_max<<<edgeBlocks, 256, 0, stream>>>(m_buf, src, dst, (unsigned int*)agg);
    combine_gemm<true><<<nTiles, 256, 0, stream>>>(in_feat, self_w1, self_b1,
                                                   agg, neigh_w1, neigh_b1, h1, 128);

    // ---- Layer 2 ----
    hipMemsetAsync(agg, 0, NODE_BYTES, stream);
    pool_gemm_relu<<<nTiles, 256, 0, stream>>>(h1, pool_w2, pool_b2, m_buf);
    edge_max<<<edgeBlocks, 256, 0, stream>>>(m_buf, src, dst, (unsigned int*)agg);
    combine_gemm<false><<<nTiles, 96, 0, stream>>>(h1, self_w2, self_b2,
                                                   agg, neigh_w2, neigh_b2, out, 40);
}